// Sinkhorn_m_70660801954104
// MI455X (gfx1250) — compile-verified
//
#include <hip/hip_runtime.h>
#include <math.h>

// Problem constants (from reference setup_inputs)
#define NB 8
#define NN 2048
#define TAU_F 1.0f
#define MAX_ITER 10
#define CONV_TOL 0.01f

typedef float v2f __attribute__((ext_vector_type(2)));
typedef float v8f __attribute__((ext_vector_type(8)));

// ---------------------------------------------------------------------------
// Exact f32 wave(32) sum-reduction using V_WMMA_F32_16X16X4_F32.
// A is 16x4 f32 (2 VGPRs): lanes 0-15 supply A[m,{0,1}], lanes 16-31 A[m,{2,3}].
// With B = ones, D[m,n] = sum_k A[m,k] = p[m] + p[m+16], broadcast across n.
// Lane L<16 then holds s1[0..7] in d[0..7]; lane L>=16 holds s1[8..15].
// One half-swap shuffle completes the full 32-lane sum (broadcast to all lanes).
// f32 MACs with a fixed accumulation order -> bit-deterministic.
// ---------------------------------------------------------------------------
__device__ __forceinline__ float wave_reduce_sum_wmma(float p) {
    v2f a; a.x = p;    a.y = 0.0f;
    v2f b; b.x = 1.0f; b.y = 1.0f;
    v8f c = {0.f, 0.f, 0.f, 0.f, 0.f, 0.f, 0.f, 0.f};
    v8f d = __builtin_amdgcn_wmma_f32_16x16x4_f32(false, a, false, b,
                                                  (short)0, c, false, false);
    float t = ((d[0] + d[1]) + (d[2] + d[3])) + ((d[4] + d[5]) + (d[6] + d[7]));
    t += __shfl_xor(t, 16, 32);
    return t;
}

__device__ __forceinline__ float wave_reduce_max(float m) {
    #pragma unroll
    for (int off = 16; off > 0; off >>= 1)
        m = fmaxf(m, __shfl_xor(m, off, 32));
    return m;
}

// ---------------------------------------------------------------------------
// Row normalization: one 128-thread block (4 waves) per row.
// Row is register-resident (4 float4 per lane) -> exactly 1 read + 1 write
// per element (plus last-buffer traffic on diff iterations).
// ---------------------------------------------------------------------------
__global__ void __launch_bounds__(128)
row_norm_kernel(const float* src, const float* last_src,
                float* dst, float* last_dst,
                float* partials, const float* __restrict__ row_prob,
                const int* done, float scale)
{
    if (done && *done) return;

    const size_t row = blockIdx.x;                 // 0 .. NB*NN-1
    const int t = threadIdx.x;
    const int wave = t >> 5, lane = t & 31;
    const size_t base = row * (size_t)NN;

    float4 x[4];
    float m = -INFINITY;
    #pragma unroll
    for (int j = 0; j < 4; ++j) {
        x[j] = *(const float4*)(src + base + 4 * t + 512 * j);
        x[j].x *= scale; x[j].y *= scale; x[j].z *= scale; x[j].w *= scale;
        m = fmaxf(m, fmaxf(fmaxf(x[j].x, x[j].y), fmaxf(x[j].z, x[j].w)));
    }

    __shared__ float sred[4];
    m = wave_reduce_max(m);
    if (lane == 0) sred[wave] = m;
    __syncthreads();
    m = fmaxf(fmaxf(sred[0], sred[1]), fmaxf(sred[2], sred[3]));
    __syncthreads();

    float p = 0.f;
    #pragma unroll
    for (int j = 0; j < 4; ++j) {
        p += __expf(x[j].x - m) + __expf(x[j].y - m);
        p += __expf(x[j].z - m) + __expf(x[j].w - m);
    }
    p = wave_reduce_sum_wmma(p);                   // v_wmma_f32_16x16x4_f32
    if (lane == 0) sred[wave] = p;
    __syncthreads();
    const float ssum = (sred[0] + sred[1]) + (sred[2] + sred[3]);

    const float lse = m + __logf(ssum);
    const float add = __logf(row_prob[row]) - lse;

    const bool diff = (partials != nullptr);
    const bool last_is_src = (last_src == src);
    float dsq = 0.f;

    #pragma unroll
    for (int j = 0; j < 4; ++j) {
        float4 v;
        v.x = x[j].x + add; if (v.x != v.x) v.x = -INFINITY;
        v.y = x[j].y + add; if (v.y != v.y) v.y = -INFINITY;
        v.z = x[j].z + add; if (v.z != v.z) v.z = -INFINITY;
        v.w = x[j].w + add; if (v.w != v.w) v.w = -INFINITY;
        if (diff) {
            float4 l = last_is_src ? x[j]
                     : *(const float4*)(last_src + base + 4 * t + 512 * j);
            float dx = v.x - l.x, dy = v.y - l.y, dz = v.z - l.z, dw = v.w - l.w;
            dsq += dx * dx + dy * dy + dz * dz + dw * dw;
            *(float4*)(last_dst + base + 4 * t + 512 * j) = v;
        }
        *(float4*)(dst + base + 4 * t + 512 * j) = v;
    }

    if (diff) {
        dsq = wave_reduce_sum_wmma(dsq);           // v_wmma again
        __syncthreads();
        if (lane == 0) sred[wave] = dsq;
        __syncthreads();
        if (t == 0) partials[row] = (sred[0] + sred[1]) + (sred[2] + sred[3]);
    }
}

// ---------------------------------------------------------------------------
// Column normalization: 256-thread block handles a 32-column stripe of one
// batch. Each wave owns one 256-row chunk of the stripe (lanes = consecutive
// columns -> coalesced 128B strided loads). Online (streaming) logsumexp
// gives 2 passes instead of 3. global_prefetch_b8 hides the 8KB stride.
// ---------------------------------------------------------------------------
#define COLT 32
__global__ void __launch_bounds__(256)
col_norm_kernel(float* buf, const float* __restrict__ col_prob, const int* done)
{
    if (done && *done) return;

    const int tiles = NN / COLT;                   // 64
    const int b = blockIdx.x / tiles;
    const int tile = blockIdx.x % tiles;
    const int lane = threadIdx.x & 31;
    const int chunk = threadIdx.x >> 5;            // 0..7
    const int c = tile * COLT + lane;
    const int rows_per_chunk = NN / 8;             // 256

    float* base = buf + ((size_t)b * NN + (size_t)chunk * rows_per_chunk) * NN + c;

    float m = -INFINITY, s = 0.f;
    for (int i = 0; i < rows_per_chunk; ++i) {
        __builtin_prefetch(base + (size_t)(i + 16) * NN, 0, 0); // global_prefetch_b8 (speculative)
        float x = base[(size_t)i * NN];
        if (x > m) { s = s * __expf(m - x) + 1.0f; m = x; }
        else       { s += __expf(x - m); }
    }

    __shared__ float sm[256], ss[256], slse[COLT];
    sm[threadIdx.x] = m;
    ss[threadIdx.x] = s;
    __syncthreads();
    if (threadIdx.x < COLT) {
        float M = sm[threadIdx.x];
        #pragma unroll
        for (int k = 1; k < 8; ++k) M = fmaxf(M, sm[threadIdx.x + 32 * k]);
        float S = 0.f;
        #pragma unroll
        for (int k = 0; k < 8; ++k)
            S += ss[threadIdx.x + 32 * k] * __expf(sm[threadIdx.x + 32 * k] - M);
        slse[threadIdx.x] = M + __logf(S);
    }
    __syncthreads();

    const float add = __logf(col_prob[(size_t)b * NN + c]) - slse[lane];
    for (int i = 0; i < rows_per_chunk; ++i) {
        float x = base[(size_t)i * NN];
        float v = x + add;
        if (v != v) v = -INFINITY;
        base[(size_t)i * NN] = v;
    }
}

// ---------------------------------------------------------------------------
// Convergence check: single block, fixed-order tree reduction (deterministic).
// ---------------------------------------------------------------------------
__global__ void __launch_bounds__(256)
check_kernel(const float* __restrict__ partials, int* done)
{
    if (*done) return;
    __shared__ float red[256];
    const int t = threadIdx.x;
    float maxd = 0.f;
    for (int b = 0; b < NB; ++b) {
        float acc = 0.f;
        for (int j = t; j < NN; j += 256) acc += partials[b * NN + j];
        red[t] = acc;
        __syncthreads();
        for (int off = 128; off > 0; off >>= 1) {
            if (t < off) red[t] += red[t + off];
            __syncthreads();
        }
        if (t == 0) maxd = fmaxf(maxd, sqrtf(red[0]));
        __syncthreads();
    }
    if (t == 0 && maxd <= CONV_TOL) *done = 1;
}

__global__ void init_done_kernel(int* done) { *done = 0; }

__global__ void __launch_bounds__(256)
exp_kernel(float* buf, size_t n4)
{
    size_t i = (size_t)blockIdx.x * blockDim.x + threadIdx.x;
    const size_t stride = (size_t)gridDim.x * blockDim.x;
    float4* p = (float4*)buf;
    for (; i < n4; i += stride) {
        float4 v = p[i];
        v.x = __expf(v.x); v.y = __expf(v.y);
        v.z = __expf(v.z); v.w = __expf(v.w);
        p[i] = v;
    }
}

// ---------------------------------------------------------------------------
extern "C" void kernel_launch(void* const* d_in, const int* in_sizes, int n_in,
                              void* d_out, int out_size, void* d_ws, size_t ws_size,
                              hipStream_t stream)
{
    (void)in_sizes; (void)n_in; (void)out_size;
    const float* s        = (const float*)d_in[0];
    const float* row_prob = (const float*)d_in[1];
    const float* col_prob = (const float*)d_in[2];
    float* out = (float*)d_out;                    // working log_s buffer, exp'd at the end

    const size_t total = (size_t)NB * NN * NN;
    const size_t need  = total * sizeof(float)        // last_even buffer
                       + (size_t)NB * NN * sizeof(float) // per-row diff partials
                       + 64;                          // done flag + slack

    const bool conv = (ws_size >= need);
    float* last     = conv ? (float*)d_ws : nullptr;
    float* partials = conv ? (last + total) : nullptr;
    int*   done     = conv ? (int*)(partials + (size_t)NB * NN) : nullptr;

    const dim3 rowGrid(NB * NN);                   // 16384 blocks x 128
    const dim3 colGrid(NB * (NN / COLT));          // 512 blocks x 256

    if (conv) init_done_kernel<<<1, 1, 0, stream>>>(done);

    // Iteration 0 (row): reads s directly (log_s0 = s/TAU), last starts as log_s0.
    row_norm_kernel<<<rowGrid, 128, 0, stream>>>(s, conv ? s : nullptr, out, last,
                                                 partials, row_prob, done, 1.0f / TAU_F);
    if (conv) check_kernel<<<1, 256, 0, stream>>>(partials, done);

    for (int i = 1; i < MAX_ITER; ++i) {
        if (i & 1) {
            col_norm_kernel<<<colGrid, 256, 0, stream>>>(out, col_prob, done);
        } else {
            row_norm_kernel<<<rowGrid, 128, 0, stream>>>(out, last, out, last,
                                                         partials, row_prob, done, 1.0f);
            if (conv) check_kernel<<<1, 256, 0, stream>>>(partials, done);
        }
    }

    exp_kernel<<<8192, 256, 0, stream>>>(out, total / 4);
}